// MSM_83545703842333
// MI455X (gfx1250) — compile-verified
//
#include <hip/hip_runtime.h>

#define NSEQ 256
#define TLEN 2000
#define KST  16

typedef float v2f __attribute__((ext_vector_type(2)));
typedef float v8f __attribute__((ext_vector_type(8)));

// lane <-> lane^16 exchange as a pure-VALU v_permlanex16_b32 (identity selects:
// each lane reads its same index from the other 16-lane row). Avoids the
// ds_bpermute LDS round-trip + s_wait_dscnt on the serial recursion chain.
__device__ __forceinline__ float sx16(float v) {
  int i = __float_as_int(v);
  int r = __builtin_amdgcn_permlanex16(i, i, 0x76543210, (int)0xfedcba98, false, false);
  return __int_as_float(r);
}

// x[r] holds value for (sequence = lane%16, state = r + 8*(lane>=16)).
// Build B (16x16 over four 4x16 slices) where B[k, n] = x_n[k], then run the
// K=16 contraction as 4 chained V_WMMA_F32_16X16X4_F32.
// D comes back in C/D layout: acc[r] = out[n = lane%16, state = r + 8*(lane>=16)].
__device__ __forceinline__ v8f step_mm(const v2f A[4], const float x[8], bool lo) {
  float s0 = sx16(x[0]); float s1 = sx16(x[1]);
  float s2 = sx16(x[2]); float s3 = sx16(x[3]);
  float s4 = sx16(x[4]); float s5 = sx16(x[5]);
  float s6 = sx16(x[6]); float s7 = sx16(x[7]);
  v2f b0, b1, b2, b3;
  // B slice layout: vgpr0 = {K=4s+0 (lanes 0-15), K=4s+2 (lanes 16-31)},
  //                 vgpr1 = {K=4s+1, K=4s+3} (mirrors documented A 16x4 layout)
  b0.x = lo ? x[0] : s2;  b0.y = lo ? x[1] : s3;   // K = 0..3
  b1.x = lo ? x[4] : s6;  b1.y = lo ? x[5] : s7;   // K = 4..7
  b2.x = lo ? s0 : x[2];  b2.y = lo ? s1 : x[3];   // K = 8..11
  b3.x = lo ? s4 : x[6];  b3.y = lo ? s5 : x[7];   // K = 12..15
  v8f acc = {0.f, 0.f, 0.f, 0.f, 0.f, 0.f, 0.f, 0.f};
  acc = __builtin_amdgcn_wmma_f32_16x16x4_f32(false, A[0], false, b0, (short)0, acc, false, false);
  acc = __builtin_amdgcn_wmma_f32_16x16x4_f32(false, A[1], false, b1, (short)0, acc, false, false);
  acc = __builtin_amdgcn_wmma_f32_16x16x4_f32(false, A[2], false, b2, (short)0, acc, false, false);
  acc = __builtin_amdgcn_wmma_f32_16x16x4_f32(false, A[3], false, b3, (short)0, acc, false, false);
  return acc;
}

__device__ __forceinline__ void store8(float* __restrict__ p, const float x[8]) {
  ((float4*)p)[0] = make_float4(x[0], x[1], x[2], x[3]);
  ((float4*)p)[1] = make_float4(x[4], x[5], x[6], x[7]);
}

__device__ __forceinline__ void load_ev8(const float* __restrict__ p, float e[8]) {
  float4 e0 = ((const float4*)p)[0];
  float4 e1 = ((const float4*)p)[1];
  e[0] = e0.x; e[1] = e0.y; e[2] = e0.z; e[3] = e0.w;
  e[4] = e1.x; e[5] = e1.y; e[6] = e1.z; e[7] = e1.w;
}

// One wave (32 threads) per block. Blocks [0,16): forward for a 16-sequence
// tile (writes scaled alpha-hat into `alpha`). Blocks [16,32): backward with
// per-step max-normalization (writes arbitrarily-scaled beta into `beta`).
__global__ __launch_bounds__(32) void hmm_fb_kernel(
    const float* __restrict__ ev, const float* __restrict__ pi,
    const float* __restrict__ Q, float* __restrict__ alpha,
    float* __restrict__ beta) {
  const int  lane = threadIdx.x;
  const bool lo   = lane < 16;
  const int  nl   = lane & 15;
  const int  hi8  = lo ? 0 : 8;
  const bool fwd  = blockIdx.x < (NSEQ / 16);
  const int  tile = fwd ? blockIdx.x : (blockIdx.x - NSEQ / 16);
  const int  seq  = tile * 16 + nl;
  const size_t rowbase = (size_t)seq * TLEN;

  // Loop-invariant A fragments (16x4 f32 layout: lanes0-15 K={4s,4s+1}, lanes16-31 K={4s+2,4s+3})
  v2f A[4];
#pragma unroll
  for (int s = 0; s < 4; ++s) {
    if (fwd) {               // A = Q^T : A[j,i] = Q[i,j]
      int i0 = 4 * s + (lo ? 0 : 2);
      A[s].x = Q[i0 * KST + nl];
      A[s].y = Q[(i0 + 1) * KST + nl];
    } else {                 // A = Q : A[i,j] = Q[i,j]
      int j0 = 4 * s + (lo ? 0 : 2);
      A[s].x = Q[nl * KST + j0];
      A[s].y = Q[nl * KST + j0 + 1];
    }
  }

  float cur[8];

  if (fwd) {
    // t = 0: p = exp(ev)*pi, renormalize
    float e[8], p[8];
    load_ev8(ev + rowbase * KST + hi8, e);
    float s = 0.f;
#pragma unroll
    for (int r = 0; r < 8; ++r) { p[r] = __expf(e[r]) * pi[hi8 + r]; s += p[r]; }
    s += sx16(s);
    float inv = __builtin_amdgcn_rcpf(s);
#pragma unroll
    for (int r = 0; r < 8; ++r) cur[r] = p[r] * inv;
    store8(alpha + rowbase * KST + hi8, cur);

    for (int t = 1; t < TLEN; ++t) {
      float ee[8];
      load_ev8(ev + (rowbase + t) * KST + hi8, ee);      // overlaps with WMMA chain
      v8f acc = step_mm(A, cur, lo);                      // acc[r] = sum_i Q[i,j] * alpha[n,i]
      float pp[8]; float z = 0.f;
#pragma unroll
      for (int r = 0; r < 8; ++r) { pp[r] = __expf(ee[r]) * acc[r]; z += pp[r]; }
      z += sx16(z);
      float iz = __builtin_amdgcn_rcpf(z);
#pragma unroll
      for (int r = 0; r < 8; ++r) cur[r] = pp[r] * iz;
      store8(alpha + (rowbase + t) * KST + hi8, cur);
    }
  } else {
#pragma unroll
    for (int r = 0; r < 8; ++r) cur[r] = 1.0f;
    store8(beta + (rowbase + (TLEN - 1)) * KST + hi8, cur);

    for (int t = TLEN - 1; t >= 1; --t) {
      float ee[8];
      load_ev8(ev + (rowbase + t) * KST + hi8, ee);
      float u[8];
#pragma unroll
      for (int r = 0; r < 8; ++r) u[r] = __expf(ee[r]) * cur[r];
      v8f acc = step_mm(A, u, lo);                        // acc[r] = sum_j Q[i,j] * u[j]
      float mx = acc[0];
#pragma unroll
      for (int r = 1; r < 8; ++r) mx = fmaxf(mx, acc[r]);
      mx = fmaxf(mx, sx16(mx));
      float im = __builtin_amdgcn_rcpf(mx);
#pragma unroll
      for (int r = 0; r < 8; ++r) cur[r] = acc[r] * im;   // per-step scale is arbitrary
      store8(beta + (rowbase + (t - 1)) * KST + hi8, cur);
    }
  }
}

// gamma = (alpha .* beta) / sum_k(alpha .* beta)   (exact: sum_k exp(la+lb) == 1)
__global__ __launch_bounds__(256) void hmm_gamma_kernel(
    float* __restrict__ g /* alpha in, gamma out */,
    const float* __restrict__ beta, int nrows) {
  int row = blockIdx.x * blockDim.x + threadIdx.x;
  if (row >= nrows) return;
  const float4* ga = (const float4*)(g + (size_t)row * KST);
  const float4* gb = (const float4*)(beta + (size_t)row * KST);
  float4 v[4]; float s = 0.f;
#pragma unroll
  for (int i = 0; i < 4; ++i) {
    float4 a = ga[i], b = gb[i];
    v[i] = make_float4(a.x * b.x, a.y * b.y, a.z * b.z, a.w * b.w);
    s += v[i].x + v[i].y + v[i].z + v[i].w;
  }
  float inv = (s > 0.f) ? __builtin_amdgcn_rcpf(s) : 0.f;
  float4* go = (float4*)(g + (size_t)row * KST);
#pragma unroll
  for (int i = 0; i < 4; ++i)
    go[i] = make_float4(v[i].x * inv, v[i].y * inv, v[i].z * inv, v[i].w * inv);
}

extern "C" void kernel_launch(void* const* d_in, const int* in_sizes, int n_in,
                              void* d_out, int out_size, void* d_ws, size_t ws_size,
                              hipStream_t stream) {
  const float* ev = (const float*)d_in[0];   // (N, T, K)
  const float* pi = (const float*)d_in[1];   // (K,)
  const float* Q  = (const float*)d_in[2];   // (K, K)
  float* gamma = (float*)d_out;              // (N, T, K); holds alpha then gamma
  float* beta  = (float*)d_ws;               // (N, T, K) scratch

  // 16 forward tiles + 16 backward tiles, one wave each, run concurrently.
  hmm_fb_kernel<<<dim3(2 * (NSEQ / 16)), dim3(32), 0, stream>>>(ev, pi, Q, gamma, beta);

  const int nrows = NSEQ * TLEN;
  hmm_gamma_kernel<<<dim3((nrows + 255) / 256), dim3(256), 0, stream>>>(gamma, beta, nrows);
}